// AtomAttention_46720654246220
// MI455X (gfx1250) — compile-verified
//
#include <hip/hip_runtime.h>
#include <cstdint>
#include <cstddef>

#define D   128
#define ED  64

typedef __attribute__((ext_vector_type(2))) float v2f;
typedef __attribute__((ext_vector_type(8))) float v8f;

// ---- ordered-int mapping for float atomicMax ----
__device__ __forceinline__ int f32_ord(float f) {
    int i = __float_as_int(f);
    return (i >= 0) ? i : (i ^ 0x7fffffff);
}
__device__ __forceinline__ float ord_f32(int i) {
    return __int_as_float((i >= 0) ? i : (i ^ 0x7fffffff));
}

// ---- CDNA5 async global->LDS copy (ASYNCcnt-tracked), 16B per lane ----
// LDS operand is the wave-relative byte offset = low 32 bits of the generic
// shared-memory address (ISA 10.2).
__device__ __forceinline__ void async_copy_b128(const float* gptr, void* lptr) {
    unsigned lds_off = (unsigned)(uintptr_t)lptr;
    asm volatile("global_load_async_to_lds_b128 %0, %1, off"
                 :: "v"(lds_off), "v"(gptr)
                 : "memory");
}
__device__ __forceinline__ void async_wait0() {
    asm volatile("s_wait_asynccnt 0x0" ::: "memory");
}

// ============================================================================
// 0) init: zero context & node_sum, node_max = -inf (ordered INT_MIN)
// ============================================================================
__global__ __launch_bounds__(256) void init_ws_k(float* __restrict__ context,
                                                 float* __restrict__ node_sum,
                                                 int* __restrict__ node_max, int N) {
    int i = blockIdx.x * 256 + threadIdx.x;
    if (i < N * D) context[i] = 0.0f;
    if (i < N * 4) { node_sum[i] = 0.0f; node_max[i] = (int)0x80000000; }
}

// ============================================================================
// 1) per-node precompute: Vnode = x @ value_w + value_b  (WMMA f32 16x16x4)
//    s_dst = x @ align_w[0:128], s_src = x @ align_w[128:256]
//    block = 256 threads (8 waves), 16 nodes per block
// ============================================================================
__global__ __launch_bounds__(256) void node_pre_k(
    const float* __restrict__ x, const float* __restrict__ align_w,
    const float* __restrict__ value_w, const float* __restrict__ value_b,
    float* __restrict__ Vnode, float* __restrict__ s_dst, float* __restrict__ s_src,
    int N) {
    __shared__ float xs[16][132];   // +4 pad -> conflict-free strided reads
    const int tid  = threadIdx.x;
    const int nb   = blockIdx.x * 16;
    const bool full = (nb + 16 <= N);   // uniform per block

    if (full) {
        // async DMA: 2 x b128 per thread, no VGPR round-trip
        #pragma unroll
        for (int p = 0; p < 2; ++p) {
            int idx = tid * 4 + p * 1024;
            int r = idx >> 7, c = idx & 127;
            async_copy_b128(x + (size_t)(nb + r) * D + c, &xs[r][c]);
        }
        async_wait0();
    } else {
        for (int i = tid; i < 16 * 128; i += 256) {
            int r = i >> 7, c = i & 127;
            int gr = nb + r;
            xs[r][c] = (gr < N) ? x[(size_t)gr * D + c] : 0.0f;
        }
    }
    __syncthreads();

    const int lane = tid & 31;
    const int wv   = tid >> 5;
    const int m    = lane & 15;
    const int hi   = lane >> 4;           // half-wave select
    const int colbase = wv * 16;
    const int n    = m;

    v8f c = {};
    #pragma unroll 4
    for (int k0 = 0; k0 < 128; k0 += 4) {
        v2f a, b;
        a.x = xs[m][k0 + 2 * hi];
        a.y = xs[m][k0 + 2 * hi + 1];
        const float* wp = value_w + (size_t)(k0 + 2 * hi) * D + colbase + n;
        b.x = wp[0];
        b.y = wp[D];
        c = __builtin_amdgcn_wmma_f32_16x16x4_f32(false, a, false, b, (short)0, c,
                                                  false, false);
    }
    const float bias = value_b[colbase + n];
    if (full) {
        #pragma unroll
        for (int r = 0; r < 8; ++r)
            Vnode[(size_t)(nb + r + 8 * hi) * D + colbase + n] = c[r] + bias;
    } else {
        #pragma unroll
        for (int r = 0; r < 8; ++r) {
            int gr = nb + r + 8 * hi;
            if (gr < N) Vnode[(size_t)gr * D + colbase + n] = c[r] + bias;
        }
    }

    // attention score halves (tiny: 8 dots of length 128 per node)
    if (tid < 128) {
        int r = tid >> 3, o = tid & 7;
        int gr = nb + r;
        if (gr < N) {
            const float* wcol = (o < 4) ? (align_w + o) : (align_w + 512 + (o - 4));
            float s = 0.0f;
            #pragma unroll 8
            for (int k = 0; k < 128; ++k) s += xs[r][k] * wcol[k * 4];
            if (o < 4) s_dst[(size_t)gr * 4 + o] = s;
            else       s_src[(size_t)gr * 4 + (o - 4)] = s;
        }
    }
}

// ============================================================================
// 2) per-edge scores + segment max (ordered-int atomicMax)
// ============================================================================
__global__ __launch_bounds__(256) void edge_scores_k(
    const int* __restrict__ ei, const float* __restrict__ edge_attr,
    const float* __restrict__ align_w, const float* __restrict__ align_b,
    const float* __restrict__ s_dst, const float* __restrict__ s_src,
    float* __restrict__ scores, int* __restrict__ node_max, int E) {
    __shared__ float lw[256];  // align_w rows 256..319 (edge part), [k*4+h]
    __shared__ float lb[4];
    const int tid = threadIdx.x;
    lw[tid] = align_w[1024 + tid];
    if (tid < 4) lb[tid] = align_b[tid];
    __syncthreads();

    int e = blockIdx.x * 256 + tid;
    if (e >= E) return;
    const int row = ei[e];
    const int col = ei[E + e];

    float s0 = 0.f, s1 = 0.f, s2 = 0.f, s3 = 0.f;
    const float4* ea = (const float4*)(edge_attr + (size_t)e * ED);
    #pragma unroll
    for (int k4 = 0; k4 < 16; ++k4) {
        float4 a = ea[k4];
        int b = k4 * 16;
        s0 += a.x * lw[b + 0] + a.y * lw[b + 4] + a.z * lw[b + 8]  + a.w * lw[b + 12];
        s1 += a.x * lw[b + 1] + a.y * lw[b + 5] + a.z * lw[b + 9]  + a.w * lw[b + 13];
        s2 += a.x * lw[b + 2] + a.y * lw[b + 6] + a.z * lw[b + 10] + a.w * lw[b + 14];
        s3 += a.x * lw[b + 3] + a.y * lw[b + 7] + a.z * lw[b + 11] + a.w * lw[b + 15];
    }
    const float* sd = s_dst + (size_t)row * 4;
    const float* ss = s_src + (size_t)col * 4;
    float4 sc;
    sc.x = s0 + sd[0] + ss[0] + lb[0];
    sc.y = s1 + sd[1] + ss[1] + lb[1];
    sc.z = s2 + sd[2] + ss[2] + lb[2];
    sc.w = s3 + sd[3] + ss[3] + lb[3];
    ((float4*)scores)[e] = sc;

    int* nm = node_max + (size_t)row * 4;
    atomicMax(&nm[0], f32_ord(sc.x));
    atomicMax(&nm[1], f32_ord(sc.y));
    atomicMax(&nm[2], f32_ord(sc.z));
    atomicMax(&nm[3], f32_ord(sc.w));
}

// ============================================================================
// 3) exp(score - max) + segment sum
// ============================================================================
__global__ __launch_bounds__(256) void exp_pass_k(
    const int* __restrict__ ei, float* __restrict__ scores,
    const int* __restrict__ node_max, float* __restrict__ node_sum, int E) {
    int e = blockIdx.x * 256 + threadIdx.x;
    if (e >= E) return;
    const int row = ei[e];
    float4 sc = ((const float4*)scores)[e];
    const int* nm = node_max + (size_t)row * 4;
    float4 w;
    w.x = __expf(sc.x - ord_f32(nm[0]));
    w.y = __expf(sc.y - ord_f32(nm[1]));
    w.z = __expf(sc.z - ord_f32(nm[2]));
    w.w = __expf(sc.w - ord_f32(nm[3]));
    ((float4*)scores)[e] = w;
    float* ns = node_sum + (size_t)row * 4;
    atomicAdd(&ns[0], w.x);
    atomicAdd(&ns[1], w.y);
    atomicAdd(&ns[2], w.z);
    atomicAdd(&ns[3], w.w);
}

// ============================================================================
// 4) weighted scatter: context[row] += alpha * Vnode[col]   (wave per edge)
// ============================================================================
__global__ __launch_bounds__(256) void scatter_k(
    const int* __restrict__ ei, const float* __restrict__ scores,
    const float* __restrict__ node_sum, const float* __restrict__ Vnode,
    float* __restrict__ context, int E) {
    int e = blockIdx.x * 8 + (threadIdx.x >> 5);
    if (e >= E) return;
    const int lane = threadIdx.x & 31;
    const int h = lane >> 3;                  // 8 lanes (32 dims) per head
    const int row = ei[e];
    const int col = ei[E + e];
    const float w = scores[(size_t)e * 4 + h];
    const float s = node_sum[(size_t)row * 4 + h];
    const float alpha = w / (s + 1e-8f);
    float4 v = ((const float4*)(Vnode + (size_t)col * D))[lane];
    float* cb = context + (size_t)row * D + lane * 4;
    atomicAdd(cb + 0, alpha * v.x);
    atomicAdd(cb + 1, alpha * v.y);
    atomicAdd(cb + 2, alpha * v.z);
    atomicAdd(cb + 3, alpha * v.w);
}

// ============================================================================
// 5) y = x + context @ proj_w + proj_b ; LayerNorm  (WMMA + shuffle reduce)
// ============================================================================
__global__ __launch_bounds__(256) void proj_ln_k(
    const float* __restrict__ x, const float* __restrict__ context,
    const float* __restrict__ proj_w, const float* __restrict__ proj_b,
    const float* __restrict__ ln_g, const float* __restrict__ ln_b,
    float* __restrict__ out, int N) {
    __shared__ float cs[16][132];
    __shared__ float ys[16][132];
    __shared__ float smu[16], srstd[16];
    const int tid  = threadIdx.x;
    const int nb   = blockIdx.x * 16;
    const bool full = (nb + 16 <= N);   // uniform per block

    if (full) {
        #pragma unroll
        for (int p = 0; p < 2; ++p) {
            int idx = tid * 4 + p * 1024;
            int r = idx >> 7, c = idx & 127;
            async_copy_b128(context + (size_t)(nb + r) * D + c, &cs[r][c]);
        }
        async_wait0();
    } else {
        for (int i = tid; i < 16 * 128; i += 256) {
            int r = i >> 7, c = i & 127;
            int gr = nb + r;
            cs[r][c] = (gr < N) ? context[(size_t)gr * D + c] : 0.0f;
        }
    }
    __syncthreads();

    const int lane = tid & 31;
    const int wv   = tid >> 5;
    const int m    = lane & 15;
    const int hi   = lane >> 4;
    const int colbase = wv * 16;
    const int n    = m;

    v8f c = {};
    #pragma unroll 4
    for (int k0 = 0; k0 < 128; k0 += 4) {
        v2f a, b;
        a.x = cs[m][k0 + 2 * hi];
        a.y = cs[m][k0 + 2 * hi + 1];
        const float* wp = proj_w + (size_t)(k0 + 2 * hi) * D + colbase + n;
        b.x = wp[0];
        b.y = wp[D];
        c = __builtin_amdgcn_wmma_f32_16x16x4_f32(false, a, false, b, (short)0, c,
                                                  false, false);
    }
    const float pb = proj_b[colbase + n];
    if (full) {
        #pragma unroll
        for (int r = 0; r < 8; ++r) {
            int rr = r + 8 * hi;
            ys[rr][colbase + n] =
                c[r] + pb + x[(size_t)(nb + rr) * D + colbase + n];
        }
    } else {
        #pragma unroll
        for (int r = 0; r < 8; ++r) {
            int rr = r + 8 * hi;
            int gr = nb + rr;
            float xv = (gr < N) ? x[(size_t)gr * D + colbase + n] : 0.0f;
            ys[rr][colbase + n] = c[r] + pb + xv;   // LDS write: no guard needed
        }
    }
    __syncthreads();

    // LayerNorm stats: 8 lanes per row, shuffle-reduce within the 8-lane group
    if (tid < 128) {
        int r = tid >> 3, seg = tid & 7;
        float s = 0.f, s2 = 0.f;
        #pragma unroll
        for (int i = 0; i < 16; ++i) {
            float v = ys[r][seg * 16 + i];
            s += v; s2 += v * v;
        }
        #pragma unroll
        for (int msk = 1; msk < 8; msk <<= 1) {
            s  += __shfl_xor(s,  msk, 32);
            s2 += __shfl_xor(s2, msk, 32);
        }
        if (seg == 0) {
            float mu  = s * (1.0f / 128.0f);
            float var = s2 * (1.0f / 128.0f) - mu * mu;
            smu[r]   = mu;
            srstd[r] = rsqrtf(var + 1e-5f);
        }
    }
    __syncthreads();

    if (full) {
        for (int i = tid; i < 16 * 128; i += 256) {
            int r = i >> 7, cc = i & 127;
            out[(size_t)(nb + r) * D + cc] =
                (ys[r][cc] - smu[r]) * srstd[r] * ln_g[cc] + ln_b[cc];
        }
    } else {
        for (int i = tid; i < 16 * 128; i += 256) {
            int r = i >> 7, cc = i & 127;
            int gr = nb + r;
            if (gr < N)
                out[(size_t)gr * D + cc] =
                    (ys[r][cc] - smu[r]) * srstd[r] * ln_g[cc] + ln_b[cc];
        }
    }
}

// ============================================================================
extern "C" void kernel_launch(void* const* d_in, const int* in_sizes, int n_in,
                              void* d_out, int out_size, void* d_ws, size_t ws_size,
                              hipStream_t stream) {
    const float* x         = (const float*)d_in[0];
    const int*   ei        = (const int*)  d_in[1];
    const float* edge_attr = (const float*)d_in[2];
    const float* align_w   = (const float*)d_in[3];
    const float* align_b   = (const float*)d_in[4];
    const float* value_w   = (const float*)d_in[5];
    const float* value_b   = (const float*)d_in[6];
    const float* proj_w    = (const float*)d_in[7];
    const float* proj_b    = (const float*)d_in[8];
    const float* ln_g      = (const float*)d_in[9];
    const float* ln_b      = (const float*)d_in[10];
    float* out = (float*)d_out;

    const int N = in_sizes[0] / D;
    const int E = in_sizes[2] / ED;

    float* ws       = (float*)d_ws;
    float* Vnode    = ws;                         // N*128
    float* context  = Vnode   + (size_t)N * D;    // N*128
    float* s_dst    = context + (size_t)N * D;    // N*4
    float* s_src    = s_dst   + (size_t)N * 4;    // N*4
    float* scores   = s_src   + (size_t)N * 4;    // E*4
    float* node_sum = scores  + (size_t)E * 4;    // N*4
    int*   node_max = (int*)(node_sum + (size_t)N * 4); // N*4

    const int nblk16 = (N + 15) / 16;

    init_ws_k<<<((size_t)N * D + 255) / 256, 256, 0, stream>>>(context, node_sum,
                                                               node_max, N);
    node_pre_k<<<nblk16, 256, 0, stream>>>(x, align_w, value_w, value_b, Vnode,
                                           s_dst, s_src, N);
    edge_scores_k<<<(E + 255) / 256, 256, 0, stream>>>(ei, edge_attr, align_w,
                                                       align_b, s_dst, s_src,
                                                       scores, node_max, E);
    exp_pass_k<<<(E + 255) / 256, 256, 0, stream>>>(ei, scores, node_max,
                                                    node_sum, E);
    scatter_k<<<(E + 7) / 8, 256, 0, stream>>>(ei, scores, node_sum, Vnode,
                                               context, E);
    proj_ln_k<<<nblk16, 256, 0, stream>>>(x, context, proj_w, proj_b, ln_g, ln_b,
                                          out, N);
}